// Projection_12438225290005
// MI455X (gfx1250) — compile-verified
//
#include <hip/hip_runtime.h>
#include <stdint.h>
#include <math.h>

// KERNEL_WIDTH = sqrt(9/pi) = 1.6925687506432687
#define KW_INV 0.59081795f          // 1 / KERNEL_WIDTH

// Fused tube-of-response projection for all three directions.
// dir 0 -> px: image viewed as transpose(2,0,1):  img[jy, k, jx]
// dir 1 -> py: image viewed as transpose(1,0,2):  img[jy, jx, k]
// dir 2 -> pz: identity:                          img[jx, jy, k]
// All are flat = jx*sx + jy*sy + k*sk on the original [D0,D1,D2] layout.
__global__ void __launch_bounds__(256)
tor_fused_kernel(const float* __restrict__ img,
                 const int*   __restrict__ grid,
                 const float* __restrict__ center,
                 const float* __restrict__ size,
                 const float* __restrict__ xl,
                 const float* __restrict__ yl,
                 const float* __restrict__ zl,
                 float*       __restrict__ out,
                 int Nx, int Ny, int Nz,
                 int bx, int by)
{
    __shared__ float slors[256 * 6];

    // ---- segment selection (uniform per block) ----
    const int b = blockIdx.x;
    int dir, lb, N, outBase;
    const float* lors;
    if (b < bx)            { dir = 0; lb = b;           N = Nx; lors = xl; outBase = 0;       }
    else if (b < bx + by)  { dir = 1; lb = b - bx;      N = Ny; lors = yl; outBase = Nx;      }
    else                   { dir = 2; lb = b - bx - by; N = Nz; lors = zl; outBase = Nx + Ny; }

    const int tid  = lb * 256 + (int)threadIdx.x;   // local LOR index in this segment
    const int lane = threadIdx.x & 31;
    const int wave = threadIdx.x >> 5;

    // ---- Stage this wave's 32 LOR rows (768 B = 48 x 16B) into LDS with
    // CDNA5 async global->LDS copies (ASYNCcnt; per-wave, no barrier needed).
    {
        const int  waveRow0   = lb * 256 + wave * 32;
        const long validBytes = ((long)N - (long)waveRow0) * 24;
        const char* gbase = (const char*)lors + (long)waveRow0 * 24;
        const unsigned ldsBase = (unsigned)(size_t)&slors[wave * 32 * 6];
        #pragma unroll
        for (int c = 0; c < 2; ++c) {
            const int chunk = lane + c * 32;
            const long off = (long)chunk * 16;
            if (chunk < 48 && off < validBytes) {
                unsigned la = ldsBase + (unsigned)off;
                uint64_t ga = (uint64_t)(gbase + off);
                asm volatile("global_load_async_to_lds_b128 %0, %1, off"
                             :: "v"(la), "v"(ga) : "memory");
            }
        }
        asm volatile("s_wait_asynccnt 0" ::: "memory");
    }

    if (tid >= N) return;

    // ---- geometry (uniform scalars) ----
    const int pa0 = (dir == 0) ? 2 : ((dir == 1) ? 1 : 0);
    const int pa1 = (dir == 2) ? 1 : 0;
    const int pa2 = (dir == 0) ? 1 : 2;

    const int   n0 = grid[pa0], n1 = grid[pa1], n2 = grid[pa2];
    const float s0 = size[pa0], s1 = size[pa1], s2 = size[pa2];
    const float vs0 = s0 / (float)n0, vs1 = s1 / (float)n1, vs2 = s2 / (float)n2;
    const float lo0 = center[pa0] - 0.5f * s0;
    const float lo1 = center[pa1] - 0.5f * s1;
    const float lo2 = center[pa2] - 0.5f * s2;
    const float inv_vs0 = 1.0f / vs0, inv_vs1 = 1.0f / vs1;

    const int D1 = grid[1], D2 = grid[2];
    int sx, sy, sk;
    if (dir == 0)      { sx = 1;       sy = D1 * D2; sk = D2; }
    else if (dir == 1) { sx = D2;      sy = D1 * D2; sk = 1;  }
    else               { sx = D1 * D2; sy = D2;      sk = 1;  }

    // ---- per-LOR setup ----
    const float* my = &slors[threadIdx.x * 6];
    const float p1x = my[0], p1y = my[1], p1z = my[2];
    const float dx = my[3] - p1x, dy = my[4] - p1y, dz = my[5] - p1z;

    const float length = __builtin_amdgcn_sqrtf(dx * dx + dy * dy + dz * dz);
    const float inv_dz = 1.0f / dz;
    const float step   = vs2 * length / fabsf(dz);

    float total = 0.0f;

    for (int k = 0; k < n2; ++k) {
        const float zc = lo2 + ((float)k + 0.5f) * vs2;
        const float t  = (zc - p1z) * inv_dz;
        const float cx = fmaf(t, dx, p1x);
        const float cy = fmaf(t, dy, p1y);
        const float fx = (cx - lo0) * inv_vs0 - 0.5f;
        const float fy = (cy - lo1) * inv_vs1 - 0.5f;
        const float rx = rintf(fx);              // v_rndne_f32 == jnp.round
        const float ry = rintf(fy);
        const int ix = (int)rx, iy = (int)ry;

        float acc = 0.0f;

        if ((ix >= 2) & (ix < n0 - 2) & (iy >= 2) & (iy < n1 - 2)) {
            // Interior fast path: fully branchless. w = max(0,...) makes
            // zero-weight taps contribute exactly 0 (matches reference).
            const float bxf = rx - fx;           // ddx at o0 = 0
            const float byf = ry - fy;
            const int base = ix * sx + iy * sy + k * sk;
            #pragma unroll
            for (int o0 = -2; o0 <= 2; ++o0) {
                const float ddx = bxf + (float)o0;
                const float dx2 = ddx * ddx;
                #pragma unroll
                for (int o1 = -2; o1 <= 2; ++o1) {
                    // corner taps: min dist 2.12 > kw -> always zero weight
                    if (o0 * o0 + o1 * o1 == 8) continue;
                    const float ddy = byf + (float)o1;
                    const float d2 = fmaf(ddy, ddy, dx2);
                    const float w = fmaxf(0.0f,
                        fmaf(__builtin_amdgcn_sqrtf(d2), -KW_INV, 1.0f));
                    acc = fmaf(w, img[base + o0 * sx + o1 * sy], acc);
                }
            }
        } else {
            // Boundary path (rare): per-tap bounds checks.
            #pragma unroll
            for (int o0 = -2; o0 <= 2; ++o0) {
                #pragma unroll
                for (int o1 = -2; o1 <= 2; ++o1) {
                    if (o0 * o0 + o1 * o1 == 8) continue;
                    const int jx = ix + o0, jy = iy + o1;
                    if ((jx >= 0) & (jx < n0) & (jy >= 0) & (jy < n1)) {
                        const float ddx = (float)jx - fx;
                        const float ddy = (float)jy - fy;
                        const float d2 = fmaf(ddy, ddy, ddx * ddx);
                        const float w = fmaxf(0.0f,
                            fmaf(__builtin_amdgcn_sqrtf(d2), -KW_INV, 1.0f));
                        acc = fmaf(w, img[jx * sx + jy * sy + k * sk], acc);
                    }
                }
            }
        }

        total += acc;
    }

    out[outBase + tid] = total * step;
}

extern "C" void kernel_launch(void* const* d_in, const int* in_sizes, int n_in,
                              void* d_out, int out_size, void* d_ws, size_t ws_size,
                              hipStream_t stream) {
    const float* img    = (const float*)d_in[0];
    const int*   grid   = (const int*)  d_in[1];
    const float* center = (const float*)d_in[2];
    const float* size   = (const float*)d_in[3];
    const float* xlors  = (const float*)d_in[4];
    const float* ylors  = (const float*)d_in[5];
    const float* zlors  = (const float*)d_in[6];
    float* out = (float*)d_out;

    const int Nx = in_sizes[4] / 6;
    const int Ny = in_sizes[5] / 6;
    const int Nz = in_sizes[6] / 6;

    const int bx = (Nx + 255) / 256;
    const int by = (Ny + 255) / 256;
    const int bz = (Nz + 255) / 256;

    tor_fused_kernel<<<dim3(bx + by + bz), dim3(256), 0, stream>>>(
        img, grid, center, size, xlors, ylors, zlors, out,
        Nx, Ny, Nz, bx, by);
}